// test_LSTM_50457275793918
// MI455X (gfx1250) — compile-verified
//
#include <hip/hip_runtime.h>

// LSTM recurrence, persistent fused kernel for MI455X (gfx1250).
//   32 workgroups x 256 threads (8 wave32's). WG w owns hidden units [16w,16w+16).
//   Wave (gate q, kgrp g): computes gate q's partial over K-half g.
//   Combined weight slice [W_ih | W_hh] (64 rows x 1024 K fp32 = 256 KB) in LDS,
//   shared by both K-groups. Per step each wave runs a 32x16x512 GEMM via
//   v_wmma_f32_16x16x4_f32; partials summed in the elementwise stage.
//   c in registers, h ping-pong in d_ws, grid barrier via atomic gen counter.

#define EMBED    512
#define HIDDEN   512
#define BATCH    32
#define SEQ      2048
#define NWG      32
#define HPW      16      // hidden units per workgroup
#define NTHREADS 256     // 8 waves: gate = wave&3, kgrp = wave>>2

typedef __attribute__((ext_vector_type(2))) float v2f;
typedef __attribute__((ext_vector_type(8))) float v8f;

__device__ __forceinline__ v8f wmma_f32_16x16x4(v2f a, v2f b, v8f c) {
  return __builtin_amdgcn_wmma_f32_16x16x4_f32(false, a, false, b, (short)0, c,
                                               false, false);
}

__device__ __forceinline__ float sigmoidf(float x) {
  return 1.0f / (1.0f + __expf(-x));
}

__launch_bounds__(NTHREADS, 1)
__global__ void lstm_persistent(const float* __restrict__ embed,  // [SEQ,B,E]
                                const float* __restrict__ W_ih,   // [4H,E]
                                const float* __restrict__ W_hh,   // [4H,H]
                                const float* __restrict__ b_ih,   // [4H]
                                const float* __restrict__ b_hh,   // [4H]
                                float* __restrict__ out,          // [S,B,H]+h+c
                                float* __restrict__ h_buf,        // [2][B,H]
                                unsigned* __restrict__ counter)   // grid barrier
{
  // Weights as K-pairs: lds_w[gate][k/2][n] = {W[row(n)][k], W[row(n)][k+1]}
  __shared__ v2f   lds_w[4][512][HPW];       // 256 KB (combined K = 1024)
  __shared__ float lds_g[2][4][BATCH][HPW];  //  16 KB partial gate tiles

  const int wg   = blockIdx.x;
  const int tid  = threadIdx.x;
  const int wave = tid >> 5;
  const int gate = wave & 3;   // i, f, g, o
  const int kgrp = wave >> 2;  // K half: 0 or 1
  const int lane = tid & 31;
  const int n15  = lane & 15;  // tile column
  const int hi   = lane >> 4;  // half-wave K sub-pair select

  // ---- one-time: stage this WG's weight slice into LDS (2 waves per gate) ----
  {
    const int rowbase = gate * HIDDEN + wg * HPW;
    for (int idx = kgrp * 32 + lane; idx < 512 * HPW; idx += 64) {
      const int kp = idx >> 4;   // k-pair 0..511
      const int n  = idx & 15;
      const int r  = rowbase + n;
      const int k0 = kp * 2;
      v2f w;
      if (k0 < EMBED) {
        w.x = W_ih[(size_t)r * EMBED + k0];
        w.y = W_ih[(size_t)r * EMBED + k0 + 1];
      } else {
        w.x = W_hh[(size_t)r * HIDDEN + (k0 - EMBED)];
        w.y = W_hh[(size_t)r * HIDDEN + (k0 - EMBED) + 1];
      }
      lds_w[gate][kp][n] = w;
    }
  }

  // ---- per-thread fused biases + cell state (2 elements each) ----
  float bi[2], bf[2], bg[2], bo[2], cc[2], hlast[2];
#pragma unroll
  for (int j = 0; j < 2; ++j) {
    const int e   = tid + NTHREADS * j;    // 0..511 = m*16 + n
    const int col = wg * HPW + (e & 15);
    bi[j] = b_ih[0 * HIDDEN + col] + b_hh[0 * HIDDEN + col];
    bf[j] = b_ih[1 * HIDDEN + col] + b_hh[1 * HIDDEN + col];
    bg[j] = b_ih[2 * HIDDEN + col] + b_hh[2 * HIDDEN + col];
    bo[j] = b_ih[3 * HIDDEN + col] + b_hh[3 * HIDDEN + col];
    cc[j] = 0.0f;
    hlast[j] = 0.0f;
  }
  __syncthreads();

  const int KK = EMBED / 8;  // 64 kk iterations per K-half per matrix

  for (int t = 0; t < SEQ; ++t) {
    v8f acc0 = {0.f, 0.f, 0.f, 0.f, 0.f, 0.f, 0.f, 0.f};  // batch 0..15
    v8f acc1 = acc0;                                      // batch 16..31

    // ---- x-contribution (no cross-WG dependency; overlaps others' step t-1)
    const float* xb  = embed + (size_t)t * BATCH * EMBED;
    const float* xr0 = xb + (size_t)n15 * EMBED + 2 * hi;
    const float* xr1 = xb + (size_t)(16 + n15) * EMBED + 2 * hi;
#pragma unroll 8
    for (int kk2 = 0; kk2 < KK; ++kk2) {
      const int kk = kgrp * KK + kk2;
      const v2f b  = lds_w[gate][2 * kk + hi][n15];
      const v2f a0 = *(const v2f*)(xr0 + 4 * kk);
      const v2f a1 = *(const v2f*)(xr1 + 4 * kk);
      acc0 = wmma_f32_16x16x4(a0, b, acc0);
      acc1 = wmma_f32_16x16x4(a1, b, acc1);
    }

    // ---- prefetch next timestep's embed block (overlaps barrier wait)
    if (t + 1 < SEQ) {
      const char* nb = (const char*)(embed + (size_t)(t + 1) * BATCH * EMBED);
#pragma unroll
      for (int p = 0; p < 2; ++p)  // 512 x 128B lines = 64 KB
        __builtin_prefetch(nb + (size_t)(p * NTHREADS + tid) * 128, 0, 3);
    }

    // ---- wait: every WG must have published h_t (NWG arrivals per step)
    if (tid == 0) {
      const unsigned target = (unsigned)(NWG * t);
      while (__hip_atomic_load(counter, __ATOMIC_ACQUIRE,
                               __HIP_MEMORY_SCOPE_AGENT) < target) {
        __builtin_amdgcn_s_sleep(2);
      }
    }
    __syncthreads();
    __threadfence();   // acquire in EVERY wave (both CU L0s see fresh h_buf)

    // ---- h-contribution
    const float* hb  = h_buf + (size_t)(t & 1) * BATCH * HIDDEN;
    const float* hr0 = hb + (size_t)n15 * HIDDEN + 2 * hi;
    const float* hr1 = hb + (size_t)(16 + n15) * HIDDEN + 2 * hi;
#pragma unroll 8
    for (int kk2 = 0; kk2 < KK; ++kk2) {
      const int kk = kgrp * KK + kk2;
      const v2f b  = lds_w[gate][EMBED / 2 + 2 * kk + hi][n15];
      const v2f a0 = *(const v2f*)(hr0 + 4 * kk);
      const v2f a1 = *(const v2f*)(hr1 + 4 * kk);
      acc0 = wmma_f32_16x16x4(a0, b, acc0);
      acc1 = wmma_f32_16x16x4(a1, b, acc1);
    }

    // ---- publish partial gate tiles (C/D layout: vgpr r -> M=r / M=r+8)
#pragma unroll
    for (int r = 0; r < 8; ++r) {
      lds_g[kgrp][gate][r + 8 * hi][n15]      = acc0[r];
      lds_g[kgrp][gate][16 + r + 8 * hi][n15] = acc1[r];
    }
    __syncthreads();

    // ---- elementwise cell update: 512 elements, 2 per thread
    float* obase = out + (size_t)t * BATCH * HIDDEN;
    float* hn    = h_buf + (size_t)((t + 1) & 1) * BATCH * HIDDEN;
#pragma unroll
    for (int j = 0; j < 2; ++j) {
      const int e = tid + NTHREADS * j;
      const int m = e >> 4;
      const int n = e & 15;
      const float ig = sigmoidf(lds_g[0][0][m][n] + lds_g[1][0][m][n] + bi[j]);
      const float fg = sigmoidf(lds_g[0][1][m][n] + lds_g[1][1][m][n] + bf[j]);
      const float gg = tanhf(lds_g[0][2][m][n] + lds_g[1][2][m][n] + bg[j]);
      const float og = sigmoidf(lds_g[0][3][m][n] + lds_g[1][3][m][n] + bo[j]);
      cc[j] = fg * cc[j] + ig * gg;
      const float hv  = og * tanhf(cc[j]);
      hlast[j]        = hv;
      const int   col = wg * HPW + n;
      obase[(size_t)m * HIDDEN + col] = hv;
      hn[(size_t)m * HIDDEN + col]    = hv;
    }

    __threadfence();   // release h writes to device scope
    __syncthreads();   // whole WG done before arrive
    if (tid == 0) atomicAdd(counter, 1u);
  }

  // ---- final (h, c) tuple outputs ----
#pragma unroll
  for (int j = 0; j < 2; ++j) {
    const int e   = tid + NTHREADS * j;
    const int m   = e >> 4;
    const int col = wg * HPW + (e & 15);
    out[(size_t)SEQ * BATCH * HIDDEN + (size_t)m * HIDDEN + col] = hlast[j];
    out[(size_t)SEQ * BATCH * HIDDEN + (size_t)BATCH * HIDDEN +
        (size_t)m * HIDDEN + col] = cc[j];
  }
}

extern "C" void kernel_launch(void* const* d_in, const int* in_sizes, int n_in,
                              void* d_out, int out_size, void* d_ws,
                              size_t ws_size, hipStream_t stream) {
  (void)in_sizes; (void)n_in; (void)out_size; (void)ws_size;
  // inputs: 0=input(int64, unused) 1=input_embed 2=W_ih 3=W_hh 4=b_ih 5=b_hh
  const float* embed = (const float*)d_in[1];
  const float* W_ih  = (const float*)d_in[2];
  const float* W_hh  = (const float*)d_in[3];
  const float* b_ih  = (const float*)d_in[4];
  const float* b_hh  = (const float*)d_in[5];
  float* out = (float*)d_out;

  // workspace: [2][B][H] fp32 h ping-pong (64 KB x 2) + barrier counter
  const size_t hbytes = (size_t)2 * BATCH * HIDDEN * sizeof(float);
  float*    h_buf   = (float*)d_ws;
  unsigned* counter = (unsigned*)((char*)d_ws + hbytes);
  hipMemsetAsync(d_ws, 0, hbytes + 64, stream);  // h0 = 0, counter = 0

  lstm_persistent<<<NWG, NTHREADS, 0, stream>>>(embed, W_ih, W_hh, b_ih, b_hh,
                                                out, h_buf, counter);
}